// ReftModule_51410758533748
// MI455X (gfx1250) — compile-verified
//
#include <hip/hip_runtime.h>

typedef float v2f __attribute__((ext_vector_type(2)));
typedef float v4f __attribute__((ext_vector_type(4)));
typedef float v8f __attribute__((ext_vector_type(8)));

constexpr int B_ = 8, L_ = 4096, D_ = 2048, P_ = 4, S_ = 4, R_ = 32;
constexpr int NTHREADS = 128;   // adapter kernel block size (4 wave32 waves)

// ---------------------------------------------------------------------------
// Kernel 1: bulk streaming copy of hidden_states -> out (256 MB each way).
// Roofline-dominant part (~22us at 23.3 TB/s). 128-bit NT loads/stores:
// the 256MB tensor exceeds the 192MB L2 and is never re-read.
// ---------------------------------------------------------------------------
__global__ __launch_bounds__(256) void reft_copy_kernel(const v4f* __restrict__ src,
                                                        v4f* __restrict__ dst,
                                                        size_t n4) {
  size_t i = (size_t)blockIdx.x * blockDim.x + threadIdx.x;
  const size_t stride = (size_t)gridDim.x * blockDim.x;
  for (; i < n4; i += stride) {
    v4f v = __builtin_nontemporal_load(&src[i]);
    __builtin_nontemporal_store(v, &dst[i]);
  }
}

// ---------------------------------------------------------------------------
// Kernel 2: per-token-slot rank-32 update via v_wmma_f32_16x16x4_f32.
// One block per slot n (8 blocks), 128 threads = 4 waves (wave32).
//
// Phase 1:  C[p][ntile] = A(8x2048) @ W_p(2048 x 16cols)  (K=2048, 512 WMMAs)
//           wave w handles p = w>>1 (0=proj,1=src), ntile = w&1.
// Phase 2:  out_rows = A + delta(8x32) @ Wp^T(32x2048); each wave covers 512
//           of the 2048 columns, 8 WMMA K-steps per 16-col tile.
//
// Padding rows 8..15 of the A operand map only to D rows 8..15 (discarded),
// so padding lanes load a *clamped* valid row instead of branching -> no
// exec-mask toggling in the WMMA loops.
//
// WMMA f32 16x16x4 fragment mapping (ISA 7.12.2), lane = {m16 = lane&15,
// hi = lane>>4}:
//   A vgpr j : A[m16, K + j + 2*hi]
//   B vgpr j : B[K + j + 2*hi, n16]
//   D vgpr r : lanes 0-15 -> row r, lanes 16-31 -> row 8+r (padding, dropped)
// ---------------------------------------------------------------------------
__global__ __launch_bounds__(NTHREADS) void reft_adapter_kernel(
    const float* __restrict__ hs,
    const float* __restrict__ pw,
    const float* __restrict__ sw,
    const float* __restrict__ sb,
    const int* __restrict__ sl,
    float* __restrict__ out) {
  const int n    = blockIdx.x;      // token slot 0..7
  const int tid  = threadIdx.x;
  const int lane = tid & 31;
  const int w    = tid >> 5;        // wave 0..3
  const int m16  = lane & 15;
  const int hi   = lane >> 4;
  const int mrow = m16 & 7;         // clamped batch row for padding lanes

  __shared__ int   s_row[B_];
  __shared__ float sC[2][B_][R_];     // [proj/src][b][o]
  __shared__ float sDelta[B_][R_];

  if (tid < B_) {
    s_row[tid] = (n < P_) ? n : (sl[tid] - S_ + (n - P_));
  }
  __syncthreads();

  // ------------------- Phase 1: A @ Wp and A @ Ws -------------------
  {
    const int p     = w >> 1;       // 0 = proj, 1 = src
    const int ntile = w & 1;        // output cols [ntile*16, ntile*16+16)
    const float* W  = (p == 0 ? pw : sw) + (size_t)n * D_ * R_;  // (2048,32)
    const int col   = ntile * 16 + m16;       // lane's N index for B operand

    // Every lane gets a valid row pointer (rows 8..15 are clamped; their
    // products land in discarded D rows).
    const float* arow = hs + ((size_t)mrow * L_ + s_row[mrow]) * D_ + 2 * hi;
    const float* wcol = W + (size_t)(2 * hi) * R_ + col;

    v8f c = {};
    #pragma unroll 8
    for (int K = 0; K < D_; K += 4) {
      v2f a;
      a.x = arow[K];
      a.y = arow[K + 1];
      v2f bm;
      bm.x = wcol[(size_t)K * R_];
      bm.y = wcol[(size_t)K * R_ + R_];
      c = __builtin_amdgcn_wmma_f32_16x16x4_f32(false, a, false, bm,
                                                (short)0, c, false, false);
    }
    if (lane < 16) {
      #pragma unroll
      for (int r = 0; r < 8; ++r) sC[p][r][ntile * 16 + lane] = c[r];
    }
  }
  __syncthreads();

  // delta = relu(src + bias) - proj  (256 elements, 128 threads: exactly 2
  // fixed iterations -> no runtime blockDim division sequence)
  #pragma unroll
  for (int it = 0; it < 2; ++it) {
    const int t = tid + it * NTHREADS;
    const int b = t >> 5, o = t & 31;
    float s = sC[1][b][o] + sb[o];
    s = s > 0.f ? s : 0.f;
    sDelta[b][o] = s - sC[0][b][o];
  }
  __syncthreads();

  // ------------- Phase 2: out_rows = A + delta @ Wp^T -------------
  {
    const float* Wp = pw + (size_t)n * D_ * R_;

    // Pre-build the 8 A-operand fragments (K = o dimension, 32 total).
    // Padding lanes read the clamped row; harmless (discarded D rows).
    v2f af[8];
    #pragma unroll
    for (int ks = 0; ks < 8; ++ks) {
      const int o = ks * 4 + 2 * hi;
      af[ks].x = sDelta[mrow][o];
      af[ks].y = sDelta[mrow][o + 1];
    }

    const int i0beg = w * (D_ / 4);             // 512 columns per wave
    for (int i0 = i0beg; i0 < i0beg + D_ / 4; i0 += 16) {
      v8f c = {};
      // B[k=o, n=i] = Wp[i, o]  (Wp row-major (D, R)); contiguous f32 pair.
      const float* wbase = Wp + (size_t)(i0 + m16) * R_ + 2 * hi;
      #pragma unroll
      for (int ks = 0; ks < 8; ++ks) {
        v2f bm;
        bm.x = wbase[ks * 4];
        bm.y = wbase[ks * 4 + 1];
        c = __builtin_amdgcn_wmma_f32_16x16x4_f32(false, af[ks], false, bm,
                                                  (short)0, c, false, false);
      }
      if (lane < 16) {
        #pragma unroll
        for (int r = 0; r < 8; ++r) {
          // Source row == destination row for every adapted token.
          const size_t idx = ((size_t)r * L_ + s_row[r]) * D_ + i0 + lane;
          out[idx] = hs[idx] + c[r];
        }
      }
    }
  }
}

// ---------------------------------------------------------------------------
extern "C" void kernel_launch(void* const* d_in, const int* /*in_sizes*/,
                              int /*n_in*/, void* d_out, int /*out_size*/,
                              void* /*d_ws*/, size_t /*ws_size*/,
                              hipStream_t stream) {
  const float* hs = (const float*)d_in[0];   // (8, 4096, 2048) f32
  const float* pw = (const float*)d_in[1];   // (8, 2048, 32)   f32
  const float* sw = (const float*)d_in[2];   // (8, 2048, 32)   f32
  const float* sb = (const float*)d_in[3];   // (32,)           f32
  const int*   sl = (const int*)d_in[4];     // (8,)            i32
  float*      out = (float*)d_out;           // (8, 4096, 2048) f32

  const size_t n4 = (size_t)B_ * L_ * D_ / 4;   // 16,777,216 float4s
  reft_copy_kernel<<<16384, 256, 0, stream>>>((const v4f*)hs, (v4f*)out, n4);
  // Runs after the copy (same stream); overwrites the 64 adapted rows.
  reft_adapter_kernel<<<P_ + S_, NTHREADS, 0, stream>>>(hs, pw, sw, sb, sl, out);
}